// TransformerDecoder_90537910600293
// MI455X (gfx1250) — compile-verified
//
#include <hip/hip_runtime.h>
#include <math.h>

#define BS 16
#define LQ 300
#define D 256
#define NH 8
#define HD 32
#define NLAYERS 6
#define DFF 1024
#define NLVL 4
#define NPTS 4
#define SUMP 16
#define LEN_V 8500
#define NROWS (BS * LQ)              // 4800
#define QT 32
#define QTILES ((LQ + QT - 1) / QT)  // 10
#define KPAD 304                     // 300 padded to multiple of 16
#define QSTRIDE 36                   // Q tile LDS stride (8B aligned, conflict-free)

typedef float v2f __attribute__((ext_vector_type(2)));
typedef float v8f __attribute__((ext_vector_type(8)));

#define V8F_ZERO ((v8f){0.f, 0.f, 0.f, 0.f, 0.f, 0.f, 0.f, 0.f})

static __device__ __forceinline__ v8f wmma_f32(v2f a, v2f b, v8f c) {
    return __builtin_amdgcn_wmma_f32_16x16x4_f32(false, a, false, b, (short)0, c, false, false);
}

// ---------------------------------------------------------------------------
// GEMM via V_WMMA_F32_16X16X4_F32:  C[n][m] = act( sum_k A[n][k]*W[m][k] + b[m] )
// A: (N x K) row-major, W: (M x K) row-major (computes A @ W^T + bias).
// One wave computes a 16x64 tile (4 accumulators share one A fragment per
// k-step -> 1.25 loads per WMMA). 8 waves / block. ACT: 0=none,1=relu,2=sigmoid
// ---------------------------------------------------------------------------
template <int ACT>
__global__ __launch_bounds__(256) void gemm_wmma_f32(
    const float* __restrict__ A, const float* __restrict__ W,
    const float* __restrict__ bias, float* __restrict__ Cout,
    int K, int M, int tilesM64, int tilesTotal)
{
    int lane = threadIdx.x & 31;
    int wv   = threadIdx.x >> 5;
    int tile = blockIdx.x * 8 + wv;
    if (tile >= tilesTotal) return;          // whole-wave uniform exit
    int tm = tile % tilesM64, tn = tile / tilesM64;
    int row0 = tn * 16, col0 = tm * 64;
    int l16 = lane & 15, hi = lane >> 4;

    const float* Ap = A + (size_t)(row0 + l16) * K + hi * 2;
    const float* Wp[4];
#pragma unroll
    for (int s = 0; s < 4; ++s) {
        Wp[s] = W + (size_t)(col0 + s * 16 + l16) * K + hi * 2;
        __builtin_prefetch(Wp[s], 0, 3);     // global_prefetch toward WGP
    }
    __builtin_prefetch(Ap, 0, 3);

    v8f acc[4];
#pragma unroll
    for (int s = 0; s < 4; ++s) acc[s] = V8F_ZERO;

#pragma unroll 4
    for (int k0 = 0; k0 < K; k0 += 4) {
        v2f a = *(const v2f*)(Ap + k0);
#pragma unroll
        for (int s = 0; s < 4; ++s) {
            v2f b = *(const v2f*)(Wp[s] + k0);
            acc[s] = wmma_f32(a, b, acc[s]);
        }
    }

#pragma unroll
    for (int s = 0; s < 4; ++s) {
        int col = col0 + s * 16 + l16;
        float bv = bias[col];
#pragma unroll
        for (int i = 0; i < 8; ++i) {
            int row = row0 + hi * 8 + i;
            float v = acc[s][i] + bv;
            if (ACT == 1) v = fmaxf(v, 0.f);
            else if (ACT == 2) v = 1.f / (1.f + expf(-v));
            Cout[(size_t)row * M + col] = v;
        }
    }
}

// ---------------------------------------------------------------------------
// MHA core, WMMA throughout: scores = Q K^T / sqrt(HD) (2x19 16x16 tiles,
// K-dim 32), wave-per-row softmax, O = P V (2x2 16x16 tiles, K-dim 304 with
// zero-padded probability columns). One block per (batch, head, 32-query tile).
// QK buffer rows: [Q(256) | K(256)] per token.
// ---------------------------------------------------------------------------
__global__ __launch_bounds__(256) void mha_attn(
    const float* __restrict__ QK, const float* __restrict__ V,
    float* __restrict__ O)
{
    __shared__ float sQ[QT * QSTRIDE];
    __shared__ float sS[QT * KPAD];

    int bh = blockIdx.x / QTILES;
    int qt = blockIdx.x % QTILES;
    int b = bh / NH, h = bh % NH;
    int q0 = qt * QT;
    int tid = threadIdx.x, lane = tid & 31, wv = tid >> 5;
    int l16 = lane & 15, hi = lane >> 4;

    // Q tile -> LDS (zero padded past LQ)
    for (int idx = tid; idx < QT * HD; idx += 256) {
        int qr = idx >> 5, d = idx & 31;
        float v = 0.f;
        if (q0 + qr < LQ) v = QK[(size_t)(b * LQ + q0 + qr) * 512 + h * HD + d];
        sQ[qr * QSTRIDE + d] = v;
    }
    __syncthreads();

    const float scale = 0.176776695296636893f;  // 1/sqrt(32)

    // ---- scores: S[32][304] = Q(32x32) @ K^T, 16x16 tiles on the WMMA pipe
    for (int ct = wv; ct < KPAD / 16; ct += 8) {
        int kcol0 = ct * 16;
        int tok = kcol0 + l16;
        if (tok >= LQ) tok = LQ - 1;        // clamped tail; columns zeroed later
        const float* Kp = QK + (size_t)(b * LQ + tok) * 512 + 256 + h * HD + hi * 2;
#pragma unroll
        for (int rt = 0; rt < 2; ++rt) {
            v8f acc = V8F_ZERO;
#pragma unroll
            for (int k0 = 0; k0 < HD; k0 += 4) {
                v2f a = *(const v2f*)(&sQ[(rt * 16 + l16) * QSTRIDE + k0 + hi * 2]);
                v2f bb = *(const v2f*)(Kp + k0);
                acc = wmma_f32(a, bb, acc);
            }
#pragma unroll
            for (int i = 0; i < 8; ++i)
                sS[(rt * 16 + hi * 8 + i) * KPAD + kcol0 + l16] = acc[i] * scale;
        }
    }
    __syncthreads();

    // ---- softmax per valid row (wave per row), plus zero the padded columns
    if (tid < QT * 4) {
        int r = tid >> 2, c = LQ + (tid & 3);
        sS[r * KPAD + c] = 0.f;
    }
    for (int r = wv; r < QT; r += 8) {
        if (q0 + r >= LQ) continue;  // wave-uniform
        float m = -3.4e38f;
        for (int k = lane; k < LQ; k += 32) m = fmaxf(m, sS[r * KPAD + k]);
#pragma unroll
        for (int off = 16; off > 0; off >>= 1) m = fmaxf(m, __shfl_xor(m, off, 32));
        float sum = 0.f;
        for (int k = lane; k < LQ; k += 32) {
            float e = expf(sS[r * KPAD + k] - m);
            sS[r * KPAD + k] = e;
            sum += e;
        }
#pragma unroll
        for (int off = 16; off > 0; off >>= 1) sum += __shfl_xor(sum, off, 32);
        float inv = 1.f / sum;
        for (int k = lane; k < LQ; k += 32) sS[r * KPAD + k] *= inv;
    }
    __syncthreads();

    // ---- O = P V : 2x2 tiles of 16x16, K-dim = 304 (padded cols are zero)
    if (wv < 4) {
        int rt = wv >> 1, dt = wv & 1;
        const float* Vb = V + (size_t)(b * LQ) * D + h * HD + dt * 16 + l16;
        v8f acc = V8F_ZERO;
        for (int k0 = 0; k0 < KPAD; k0 += 4) {
            v2f a = *(const v2f*)(&sS[(rt * 16 + l16) * KPAD + k0 + hi * 2]);
            int t0 = k0 + hi * 2;
            int t1 = t0 + 1;
            if (t0 > LQ - 1) t0 = LQ - 1;   // P is zero there -> no contribution
            if (t1 > LQ - 1) t1 = LQ - 1;
            v2f bb = { Vb[(size_t)t0 * D], Vb[(size_t)t1 * D] };
            acc = wmma_f32(a, bb, acc);
        }
#pragma unroll
        for (int i = 0; i < 8; ++i) {
            int qr = rt * 16 + hi * 8 + i;
            if (q0 + qr < LQ)
                O[(size_t)(b * LQ + q0 + qr) * D + h * HD + dt * 16 + l16] = acc[i];
        }
    }
}

// ---------------------------------------------------------------------------
// LayerNorm over D=256, one block (256 threads) per row
// ---------------------------------------------------------------------------
__global__ __launch_bounds__(256) void ln_kernel(
    const float* __restrict__ x, const float* __restrict__ g,
    const float* __restrict__ bta, float* __restrict__ out)
{
    __shared__ float red[256];
    int row = blockIdx.x, tid = threadIdx.x;
    float v = x[(size_t)row * D + tid];
    red[tid] = v;
    __syncthreads();
    for (int s = 128; s > 0; s >>= 1) { if (tid < s) red[tid] += red[tid + s]; __syncthreads(); }
    float mean = red[0] * (1.f / D);
    __syncthreads();
    float dv = v - mean;
    red[tid] = dv * dv;
    __syncthreads();
    for (int s = 128; s > 0; s >>= 1) { if (tid < s) red[tid] += red[tid + s]; __syncthreads(); }
    float var = red[0] * (1.f / D);
    float r = rsqrtf(var + 1e-5f);
    out[(size_t)row * D + tid] = dv * r * g[tid] + bta[tid];
}

// ---------------------------------------------------------------------------
// Elementwise helpers (float4 vectorized where shapes allow)
// ---------------------------------------------------------------------------
__global__ void add2_kernel(const float4* __restrict__ a, const float4* __restrict__ b,
                            float4* __restrict__ o, int n4)
{
    int i = blockIdx.x * 256 + threadIdx.x;
    if (i < n4) {
        float4 x = a[i], y = b[i];
        o[i] = make_float4(x.x + y.x, x.y + y.y, x.z + y.z, x.w + y.w);
    }
}

__global__ void addclip_kernel(const float4* __restrict__ a, const float4* __restrict__ b,
                               float4* __restrict__ o, int n4)
{
    int i = blockIdx.x * 256 + threadIdx.x;
    if (i < n4) {
        float4 x = a[i], y = b[i];
        float4 r;
        r.x = fminf(fmaxf(x.x + y.x, -65504.f), 65504.f);
        r.y = fminf(fmaxf(x.y + y.y, -65504.f), 65504.f);
        r.z = fminf(fmaxf(x.z + y.z, -65504.f), 65504.f);
        r.w = fminf(fmaxf(x.w + y.w, -65504.f), 65504.f);
        o[i] = r;
    }
}

__global__ void concat2_kernel(const float4* __restrict__ a, const float4* __restrict__ b,
                               float4* __restrict__ o, int n4)  // n4 = rows * 128
{
    int i = blockIdx.x * 256 + threadIdx.x;
    if (i < n4) {
        int r = i >> 7, c = i & 127;  // 128 float4 per 512-col row
        o[i] = (c < 64) ? a[(size_t)r * 64 + c] : b[(size_t)r * 64 + (c - 64)];
    }
}

__global__ void gate_combine_kernel(const float* __restrict__ t, const float* __restrict__ s,
                                    const float* __restrict__ gsig, float* __restrict__ o,
                                    int n)  // n = rows * 256
{
    int i = blockIdx.x * 256 + threadIdx.x;
    if (i < n) {
        int r = i >> 8, c = i & 255;
        o[i] = gsig[(size_t)r * 512 + c] * t[i] + gsig[(size_t)r * 512 + 256 + c] * s[i];
    }
}

__global__ void aw_softmax_kernel(float* __restrict__ aw, int total)  // total = NROWS*NH
{
    int i = blockIdx.x * 256 + threadIdx.x;
    if (i >= total) return;
    float* p = aw + (size_t)i * SUMP;
    float m = -3.4e38f;
#pragma unroll
    for (int j = 0; j < SUMP; ++j) m = fmaxf(m, p[j]);
    float e[SUMP], sum = 0.f;
#pragma unroll
    for (int j = 0; j < SUMP; ++j) { e[j] = expf(p[j] - m); sum += e[j]; }
    float inv = 1.f / sum;
#pragma unroll
    for (int j = 0; j < SUMP; ++j) p[j] = e[j] * inv;
}

// ---------------------------------------------------------------------------
// Deformable attention sampling: one wave per (token, head), lane = channel
// (every corner gather is one coalesced 128B transaction, wave-uniform control)
// ---------------------------------------------------------------------------
__global__ __launch_bounds__(256) void deform_attn_kernel(
    const float* __restrict__ value, const float* __restrict__ refp,
    const float* __restrict__ so, const float* __restrict__ aw,
    float* __restrict__ out)
{
    int lane = threadIdx.x & 31;
    int wid = blockIdx.x * 8 + (threadIdx.x >> 5);  // (n,h)
    int n = wid >> 3, h = wid & 7;
    int b = n / LQ;

    const int Hs_[4] = {80, 40, 20, 10};
    const int Ws_[4] = {80, 40, 20, 10};
    const int off_[4] = {0, 6400, 8000, 8400};

    float r0 = refp[(size_t)n * 4 + 0];
    float r1 = refp[(size_t)n * 4 + 1];
    float r2 = refp[(size_t)n * 4 + 2];
    float r3 = refp[(size_t)n * 4 + 3];

    float acc = 0.f;
    const float* vbase = value + (size_t)b * LEN_V * D + h * HD + lane;

#pragma unroll
    for (int lvl = 0; lvl < NLVL; ++lvl) {
        int Hs = Hs_[lvl], Ws = Ws_[lvl];
        const float* vlvl = vbase + (size_t)off_[lvl] * D;
#pragma unroll
        for (int p = 0; p < NPTS; ++p) {
            int sp = lvl * NPTS + p;
            float sx = so[(size_t)n * 256 + (h * SUMP + sp) * 2 + 0];
            float sy = so[(size_t)n * 256 + (h * SUMP + sp) * 2 + 1];
            float w  = aw[(size_t)n * 128 + h * SUMP + sp];
            float lx = r0 + sx * (1.f / NPTS) * r2 * 0.5f;
            float ly = r1 + sy * (1.f / NPTS) * r3 * 0.5f;
            float x = lx * (float)Ws - 0.5f;
            float y = ly * (float)Hs - 0.5f;
            float x0f = floorf(x), y0f = floorf(y);
            int x0 = (int)x0f, y0 = (int)y0f;
            int x1 = x0 + 1, y1 = y0 + 1;
            float wx1 = x - x0f, wx0 = 1.f - wx1;
            float wy1 = y - y0f, wy0 = 1.f - wy1;

            if (x0 >= 0 && x0 < Ws && y0 >= 0 && y0 < Hs)
                acc += w * wx0 * wy0 * vlvl[(size_t)(y0 * Ws + x0) * D];
            if (x1 >= 0 && x1 < Ws && y0 >= 0 && y0 < Hs)
                acc += w * wx1 * wy0 * vlvl[(size_t)(y0 * Ws + x1) * D];
            if (x0 >= 0 && x0 < Ws && y1 >= 0 && y1 < Hs)
                acc += w * wx0 * wy1 * vlvl[(size_t)(y1 * Ws + x0) * D];
            if (x1 >= 0 && x1 < Ws && y1 >= 0 && y1 < Hs)
                acc += w * wx1 * wy1 * vlvl[(size_t)(y1 * Ws + x1) * D];
        }
    }
    out[(size_t)n * D + h * HD + lane] = acc;
}

// ---------------------------------------------------------------------------
// Host side
// ---------------------------------------------------------------------------
static inline void launch_gemm(const float* A, const float* W, const float* b,
                               float* C, int K, int M, int act, hipStream_t s)
{
    int tilesM64 = M / 64;
    int tilesTotal = (NROWS / 16) * tilesM64;
    int grid = (tilesTotal + 7) / 8;
    if (act == 0)
        gemm_wmma_f32<0><<<grid, 256, 0, s>>>(A, W, b, C, K, M, tilesM64, tilesTotal);
    else if (act == 1)
        gemm_wmma_f32<1><<<grid, 256, 0, s>>>(A, W, b, C, K, M, tilesM64, tilesTotal);
    else
        gemm_wmma_f32<2><<<grid, 256, 0, s>>>(A, W, b, C, K, M, tilesM64, tilesTotal);
}

extern "C" void kernel_launch(void* const* d_in, const int* in_sizes, int n_in,
                              void* d_out, int out_size, void* d_ws, size_t ws_size,
                              hipStream_t stream)
{
    (void)in_sizes; (void)n_in; (void)out_size; (void)ws_size;

    const float* target = (const float*)d_in[0];
    const float* refp   = (const float*)d_in[1];
    const float* value  = (const float*)d_in[2];
    const float* pos    = (const float*)d_in[3];
    const float* Wqkv   = (const float*)d_in[4];
    const float* bqkv   = (const float*)d_in[5];
    const float* Wo     = (const float*)d_in[6];
    const float* bo     = (const float*)d_in[7];
    const float* n1g    = (const float*)d_in[8];
    const float* n1b    = (const float*)d_in[9];
    const float* soW    = (const float*)d_in[10];
    const float* sob    = (const float*)d_in[11];
    const float* awW    = (const float*)d_in[12];
    const float* awb    = (const float*)d_in[13];
    const float* gW     = (const float*)d_in[14];
    const float* gb     = (const float*)d_in[15];
    const float* gng    = (const float*)d_in[16];
    const float* gnb    = (const float*)d_in[17];
    const float* l1W    = (const float*)d_in[18];
    const float* l1b    = (const float*)d_in[19];
    const float* l2W    = (const float*)d_in[20];
    const float* l2b    = (const float*)d_in[21];
    const float* n3g    = (const float*)d_in[22];
    const float* n3b    = (const float*)d_in[23];

    float* ws = (float*)d_ws;
    const size_t U = (size_t)NROWS * D;  // 4800*256 floats
    float* t    = ws + 0 * U;            // running residual stream
    float* tmp  = ws + 1 * U;            // pre-LN sums
    float* bufA = ws + 2 * U;            // q / qc
    float* bufB = ws + 3 * U;            // qk / cat      (2U)
    float* bufC = ws + 5 * U;            // v / t2 / so
    float* bufD = ws + 6 * U;            // attn o / samp / ffn out
    float* bufE = ws + 7 * U;            // gates         (2U)
    float* bufG = ws + 9 * U;            // aw            (0.5U)
    float* bufH = ws + 2 * U;            // ffn hidden    (4U, aliases A,B,C)

    hipMemcpyAsync(t, target, U * sizeof(float), hipMemcpyDeviceToDevice, stream);

    const int nE  = (int)U;                   // 1,228,800
    const int n4  = nE / 4;
    const int g4  = (n4 + 255) / 256;
    const int gE  = (nE + 255) / 256;

    for (int i = 0; i < NLAYERS; ++i) {
        const float* Wqkv_i = Wqkv + (size_t)i * 3 * D * D;
        const float* bqkv_i = bqkv + (size_t)i * 3 * D;
        const float* Wo_i   = Wo   + (size_t)i * D * D;
        const float* bo_i   = bo   + (size_t)i * D;
        const float* soW_i  = soW  + (size_t)i * 256 * D;
        const float* sob_i  = sob  + (size_t)i * 256;
        const float* awW_i  = awW  + (size_t)i * 128 * D;
        const float* awb_i  = awb  + (size_t)i * 128;
        const float* gW_i   = gW   + (size_t)i * 512 * 512;
        const float* gb_i   = gb   + (size_t)i * 512;
        const float* l1W_i  = l1W  + (size_t)i * DFF * D;
        const float* l1b_i  = l1b  + (size_t)i * DFF;
        const float* l2W_i  = l2W  + (size_t)i * D * DFF;
        const float* l2b_i  = l2b  + (size_t)i * D;

        // ---- self attention ----
        add2_kernel<<<g4, 256, 0, stream>>>((const float4*)t, (const float4*)pos,
                                            (float4*)bufA, n4);               // q = t + pos
        launch_gemm(bufA, Wqkv_i,           bqkv_i,       bufB, D, 512, 0, stream); // [Q|K]
        launch_gemm(t,    Wqkv_i + 512 * D, bqkv_i + 512, bufC, D, 256, 0, stream); // V
        mha_attn<<<BS * NH * QTILES, 256, 0, stream>>>(bufB, bufC, bufD);     // heads out
        launch_gemm(bufD, Wo_i, bo_i, bufC, D, 256, 0, stream);               // t2 = o @ Wo^T
        add2_kernel<<<g4, 256, 0, stream>>>((const float4*)t, (const float4*)bufC,
                                            (float4*)tmp, n4);
        ln_kernel<<<NROWS, 256, 0, stream>>>(tmp, n1g + (size_t)i * D, n1b + (size_t)i * D, t);

        // ---- deformable attention ----
        add2_kernel<<<g4, 256, 0, stream>>>((const float4*)t, (const float4*)pos,
                                            (float4*)bufA, n4);               // qc = t + pos
        launch_gemm(bufA, soW_i, sob_i, bufC, D, 256, 0, stream);             // sampling offsets
        launch_gemm(bufA, awW_i, awb_i, bufG, D, 128, 0, stream);             // attn weights raw
        aw_softmax_kernel<<<(NROWS * NH + 255) / 256, 256, 0, stream>>>(bufG, NROWS * NH);
        deform_attn_kernel<<<NROWS * NH / 8, 256, 0, stream>>>(value, refp, bufC, bufG, bufD);

        // ---- gated residual ----
        concat2_kernel<<<(NROWS * 128 + 255) / 256, 256, 0, stream>>>(
            (const float4*)t, (const float4*)bufD, (float4*)bufB, NROWS * 128);
        launch_gemm(bufB, gW_i, gb_i, bufE, 512, 512, 2, stream);             // sigmoid gates
        gate_combine_kernel<<<gE, 256, 0, stream>>>(t, bufD, bufE, tmp, nE);
        ln_kernel<<<NROWS, 256, 0, stream>>>(tmp, gng + (size_t)i * D, gnb + (size_t)i * D, t);

        // ---- FFN ----
        launch_gemm(t,    l1W_i, l1b_i, bufH, D,   DFF, 1, stream);           // relu hidden
        launch_gemm(bufH, l2W_i, l2b_i, bufD, DFF, D,   0, stream);
        addclip_kernel<<<g4, 256, 0, stream>>>((const float4*)t, (const float4*)bufD,
                                               (float4*)tmp, n4);
        ln_kernel<<<NROWS, 256, 0, stream>>>(tmp, n3g + (size_t)i * D, n3b + (size_t)i * D, t);
    }

    hipMemcpyAsync(d_out, t, U * sizeof(float), hipMemcpyDeviceToDevice, stream);
}